// ContrastiveLoss_9191230013502
// MI455X (gfx1250) — compile-verified
//
#include <hip/hip_runtime.h>

typedef __attribute__((ext_vector_type(2))) float v2f;
typedef __attribute__((ext_vector_type(8))) float v8f;

#define LOG2E_F 1.44269504088896340736f
#define LN2_F   0.69314718055994530942f
#define C_EXP   (2.0f * LOG2E_F)      /* exp(2*p) = 2^(C_EXP*p) */
#define E2_F    7.38905609893065f     /* exp(2): diagonal term sim=1, /TEMP=2 */

#define NMAX 16384
#define ROWS_PER_BLOCK 16
#define WAVES 4
#define THREADS (WAVES * 32)

// One block handles a 16-row strip of the N x N similarity matrix.
// reps (64KB) staged once in LDS; each wave covers an interleaved set of
// 4-column groups; V_WMMA_F32_16X16X4_F32 with an all-ones B matrix performs
// the row reduction on the matrix pipe while VALU computes rcp/exp2.
__global__ __launch_bounds__(THREADS) void cl_row_kernel(
    const float* __restrict__ emb_i, const float* __restrict__ emb_j,
    float* __restrict__ partials, int B, int N)
{
    __shared__ float reps[NMAX];
    __shared__ float wsum[WAVES * ROWS_PER_BLOCK];
    __shared__ float rowloss[ROWS_PER_BLOCK];

    const int tid = threadIdx.x;

    // Stage reps = concat(emb_i, emb_j) into LDS, 16B granularity
    // (global_load_b128 + ds_store_b128).
    {
        const float4* si = (const float4*)emb_i;
        const float4* sj = (const float4*)emb_j;
        float4* dst = (float4*)reps;
        const int q = B >> 2;
        for (int i = tid; i < q; i += THREADS) dst[i] = si[i];
        for (int i = tid; i < q; i += THREADS) dst[q + i] = sj[i];
    }
    __syncthreads();

    const int lane = tid & 31;
    const int w    = tid >> 5;
    const int m    = lane & 15;          // A-matrix row for this lane
    const int k0   = (lane >> 4) << 1;   // K offset: lanes 0-15 -> {0,1}, 16-31 -> {2,3}
    const int rowBase = blockIdx.x * ROWS_PER_BLOCK;
    const float ra = reps[rowBase + m];

    v8f c = {};                          // 16x16 f32 accumulator (8 VGPRs)
    const v2f bones = {1.0f, 1.0f};      // B = ones(4x16): D[m,n] += sum_k A[m,k]

    // Thread-invariant trip count (no exec-masked loop), per-lane column
    // offset folded into the base pointer so the unroll can cluster ds_loads.
    const float* colp = &reps[w * 4 + k0];
    const int niter = N >> 4;            // 16 columns per block-round
    #pragma unroll 4
    for (int it = 0; it < niter; ++it) {
        const float rb0 = colp[it * 16];
        const float rb1 = colp[it * 16 + 1];
        const float t0 = C_EXP * __builtin_amdgcn_rcpf(1.0f + fabsf(ra - rb0));
        const float t1 = C_EXP * __builtin_amdgcn_rcpf(1.0f + fabsf(ra - rb1));
        v2f a;
        a.x = __builtin_amdgcn_exp2f(t0);   // exp(sim/TEMP) for K=k0
        a.y = __builtin_amdgcn_exp2f(t1);   // exp(sim/TEMP) for K=k0+1
        c = __builtin_amdgcn_wmma_f32_16x16x4_f32(
                /*neg_a=*/false, a, /*neg_b=*/false, bones,
                /*c_mod=*/(short)0, c, /*reuse_a=*/false, /*reuse_b=*/false);
    }

    // C/D layout: VGPR v, lanes 0-15 hold D[v][n], lanes 16-31 hold D[8+v][n];
    // all n-columns identical (B was ones), so lanes 0 and 16 carry the sums.
    if (lane == 0 || lane == 16) {
        const int mbase = (lane == 0) ? 0 : 8;
        #pragma unroll
        for (int v = 0; v < 8; ++v)
            wsum[w * ROWS_PER_BLOCK + mbase + v] = c[v];
    }
    __syncthreads();

    // Combine the 4 wave partials per row; add positive term.
    if (tid < ROWS_PER_BLOCK) {
        float S = 0.0f;
        #pragma unroll
        for (int ww = 0; ww < WAVES; ++ww) S += wsum[ww * ROWS_PER_BLOCK + tid];
        const float den = S - E2_F;                       // drop diagonal (mask = 1-eye)
        const int row  = rowBase + tid;
        const int prow = (row < B) ? (row + B) : (row - B);
        const float pos = __builtin_amdgcn_rcpf(1.0f + fabsf(reps[row] - reps[prow]));
        // loss_partial = log(den) - positives/TEMP ; log = log2 * ln2
        rowloss[tid] = __builtin_amdgcn_logf(den) * LN2_F - 2.0f * pos;
    }
    __syncthreads();

    if (tid == 0) {
        float s = 0.0f;
        #pragma unroll
        for (int r = 0; r < ROWS_PER_BLOCK; ++r) s += rowloss[r];
        partials[blockIdx.x] = s;
    }
}

// Deterministic final reduction of the per-block partials -> scalar loss.
__global__ __launch_bounds__(256) void cl_reduce_kernel(
    const float* __restrict__ partials, float* __restrict__ out,
    int nPartials, int N)
{
    __shared__ float red[256];
    const int tid = threadIdx.x;
    float s = 0.0f;
    for (int i = tid; i < nPartials; i += 256) s += partials[i];
    red[tid] = s;
    __syncthreads();
    for (int off = 128; off > 0; off >>= 1) {
        if (tid < off) red[tid] += red[tid + off];
        __syncthreads();
    }
    if (tid == 0) out[0] = red[0] / (float)N;
}

extern "C" void kernel_launch(void* const* d_in, const int* in_sizes, int n_in,
                              void* d_out, int out_size, void* d_ws, size_t ws_size,
                              hipStream_t stream) {
    const float* emb_i = (const float*)d_in[0];
    const float* emb_j = (const float*)d_in[1];
    const int B = in_sizes[0];   // 8192
    const int N = 2 * B;         // 16384
    float* out      = (float*)d_out;
    float* partials = (float*)d_ws;      // N/16 floats of scratch
    const int nBlocks = N / ROWS_PER_BLOCK;  // 1024

    cl_row_kernel<<<nBlocks, THREADS, 0, stream>>>(emb_i, emb_j, partials, B, N);
    cl_reduce_kernel<<<1, 256, 0, stream>>>(partials, out, nBlocks, N);
}